// QuantumNeuralNetwork_73083163509199
// MI455X (gfx1250) — compile-verified
//
#include <hip/hip_runtime.h>
#include <hip/hip_bf16.h>

typedef __attribute__((ext_vector_type(2))) float v2f;
typedef __attribute__((ext_vector_type(8))) float v8f;

#define NQ   14
#define NST  16384   // 2^14
#define NT   256     // threads per block (8 wave32)

__device__ __forceinline__ int par(int v) { return __popc(v) & 1; }

// --- CNOT-ring permutation (GF(2)-linear):  y_v = parity(e & mask_v),
// mask_0 = 0x3FFE, mask_v = (2<<v)-1.  Column of bit w:
__device__ __forceinline__ constexpr int pcol(int w) {
    return (w == 0) ? 0x3FFE : (1 | (0x3FFF & ~((1 << w) - 1)));
}
__device__ __forceinline__ int permMap(int v) {
    int y = 0;
#pragma unroll
    for (int w = 0; w < 14; ++w)
        y ^= ((v >> w) & 1) ? pcol(w) : 0;
    return y;
}

// U[i][j] of a fused 4-qubit RY block: U = r3 (x) r2 (x) r1 (x) r0
__device__ __forceinline__ float uelem(int i, int j, const float* cc, const float* ss) {
    float p = 1.0f;
    int neg = 0;
#pragma unroll
    for (int k = 0; k < 4; ++k) {
        int ib = (i >> k) & 1;
        int jb = (j >> k) & 1;
        p *= (ib ^ jb) ? ss[k] : cc[k];
        neg ^= (~ib & jb) & 1;
    }
    return __uint_as_float(__float_as_uint(p) ^ ((unsigned)neg << 31));
}

__device__ __forceinline__ v8f wmma4(v2f a, v2f b, v8f c) {
    return __builtin_amdgcn_wmma_f32_16x16x4_f32(
        false, a, false, b, (short)0, c, false, false);
}

// ---- Fused RY on bits 0..3 then 4..7:  per 256-elem block  M' = U2*(M*U1^T).
// T = M*U1^T lands in C/D layout; convert to B layout with one half-wave
// exchange (shfl_xor 16) + hi-selects, then multiply by U2 from A registers.
__device__ __forceinline__ void fused_lm(float* st, const float* ctab, const float* stab, int tid) {
    const int lane = tid & 31;
    const int wvs  = __builtin_amdgcn_readfirstlane(tid >> 5);
    const int ln = lane & 15, hi = lane >> 4;

    float c0r[4], s0r[4], c1r[4], s1r[4];
#pragma unroll
    for (int k = 0; k < 4; ++k) {
        c0r[k] = ctab[k];     s0r[k] = stab[k];       // wires 0..3
        c1r[k] = ctab[4 + k]; s1r[k] = stab[4 + k];   // wires 4..7
    }
    float bu[4][2], au[4][2];
#pragma unroll
    for (int k = 0; k < 4; ++k) {
        int c0 = 4 * k + 2 * hi;
        bu[k][0] = uelem(ln, c0, c0r, s0r);      // B of U1^T
        bu[k][1] = uelem(ln, c0 + 1, c0r, s0r);
        au[k][0] = uelem(ln, c0, c1r, s1r);      // A of U2
        au[k][1] = uelem(ln, c0 + 1, c1r, s1r);
    }
    for (int q = wvs; q < 64; q += 8) {          // scalar loop, EXEC full
        const int base = q << 8;
        v8f t = {0.f, 0.f, 0.f, 0.f, 0.f, 0.f, 0.f, 0.f};
#pragma unroll
        for (int k = 0; k < 4; ++k) {
            v2f a = *(const v2f*)(st + base + ln * 16 + 4 * k + 2 * hi);
            v2f b; b.x = bu[k][0]; b.y = bu[k][1];
            t = wmma4(a, b, t);
        }
        // C-layout -> B-layout: rows needed by the other half-wave
        float xr[8];
#pragma unroll
        for (int r = 0; r < 8; ++r) xr[r] = __shfl_xor(t[r], 16, 32);
        v2f bk[4];
        bk[0].x = hi ? xr[2] : t[0];  bk[0].y = hi ? xr[3] : t[1];
        bk[1].x = hi ? xr[6] : t[4];  bk[1].y = hi ? xr[7] : t[5];
        bk[2].x = hi ? t[2] : xr[0];  bk[2].y = hi ? t[3] : xr[1];
        bk[3].x = hi ? t[6] : xr[4];  bk[3].y = hi ? t[7] : xr[5];

        v8f d = {0.f, 0.f, 0.f, 0.f, 0.f, 0.f, 0.f, 0.f};
#pragma unroll
        for (int k = 0; k < 4; ++k) {
            v2f a; a.x = au[k][0]; a.y = au[k][1];
            d = wmma4(a, bk[k], d);
        }
#pragma unroll
        for (int r = 0; r < 8; ++r)
            st[base + (r + 8 * hi) * 16 + ln] = d[r];
    }
}

// ---- Fused RY on bits 8..11 (WMMA) + RY on 12,13 (register butterfly across
// the four hh accumulators) + CNOT-ring permutation folded into the scatter.
__device__ __forceinline__ void fused_hi(const float* src, float* dst,
                                         const float* ctab, const float* stab, int tid) {
    const int lane = tid & 31;
    const int wvs  = __builtin_amdgcn_readfirstlane(tid >> 5);
    const int ln = lane & 15, hi = lane >> 4;

    float cc[4], ss[4];
#pragma unroll
    for (int k = 0; k < 4; ++k) { cc[k] = ctab[8 + k]; ss[k] = stab[8 + k]; }
    float au[4][2];
#pragma unroll
    for (int k = 0; k < 4; ++k) {
        int c0 = 4 * k + 2 * hi;
        au[k][0] = uelem(ln, c0, cc, ss);        // A of U3
        au[k][1] = uelem(ln, c0 + 1, cc, ss);
    }
    const float c12 = ctab[12], s12 = stab[12], c13 = ctab[13], s13 = stab[13];
    const int plane = permMap(ln | (hi << 11));  // per-lane part of P(e)

    for (int ct_ = wvs; ct_ < 16; ct_ += 8) {    // scalar column-tile loop
        const int psc_ct = permMap(ct_ << 4);
        v8f dv[4];
#pragma unroll
        for (int hh = 0; hh < 4; ++hh) {
            const int base = (hh << 12) + (ct_ << 4);
            v8f d = {0.f, 0.f, 0.f, 0.f, 0.f, 0.f, 0.f, 0.f};
#pragma unroll
            for (int k = 0; k < 4; ++k) {
                const int r0 = 4 * k + 2 * hi;
                v2f a; a.x = au[k][0]; a.y = au[k][1];
                v2f b; b.x = src[base + r0 * 256 + ln];
                b.y = src[base + (r0 + 1) * 256 + ln];
                d = wmma4(a, b, d);
            }
            dv[hh] = d;
        }
        // wires 12 (hh bit0) and 13 (hh bit1): butterfly across accumulators
        v8f o[4];
#pragma unroll
        for (int e = 0; e < 8; ++e) {
            float n0 = c12 * dv[0][e] - s12 * dv[1][e];
            float n1 = s12 * dv[0][e] + c12 * dv[1][e];
            float n2 = c12 * dv[2][e] - s12 * dv[3][e];
            float n3 = s12 * dv[2][e] + c12 * dv[3][e];
            o[0][e] = c13 * n0 - s13 * n2;  o[2][e] = s13 * n0 + c13 * n2;
            o[1][e] = c13 * n1 - s13 * n3;  o[3][e] = s13 * n1 + c13 * n3;
        }
        // scatter, permuted: e = hh*4096 + (r+8hi)*256 + ct*16 + ln
#pragma unroll
        for (int hh = 0; hh < 4; ++hh) {
            const int psc = psc_ct ^ permMap(hh << 12);   // folds per hh
#pragma unroll
            for (int r = 0; r < 8; ++r)
                dst[plane ^ psc ^ permMap(r << 8)] = o[hh][r];
        }
    }
}

__global__ __launch_bounds__(NT) void QuantumNeuralNetwork_73083163509199_kernel(
    const float* __restrict__ x, const float* __restrict__ wgt, float* __restrict__ out) {

    __shared__ float sbuf0[NST];
    __shared__ float sbuf1[NST];
    __shared__ float saux[56];    // [0..13] cos((x+th0)/2) [14..27] sin((x+th0)/2)
                                  // [28..41] cos(th1/2)    [42..55] sin(th1/2)
    __shared__ float sred[8 * NQ];

    const int tid = threadIdx.x;
    const int b   = blockIdx.x;
    const int lane = tid & 31, wv = tid >> 5;

    // --- trig tables.  RY(x) then RY(th0) on the same wire = RY(x+th0):
    // the whole layer-0 rotation sweep folds into the encoding angles.
    if (tid < NQ) {
        float a = 0.5f * (x[b * NQ + tid] + wgt[tid]);       // wgt[0][w]
        float sv, cv; sincosf(a, &sv, &cv);
        saux[tid] = cv; saux[NQ + tid] = sv;
    } else if (tid >= 32 && tid < 32 + NQ) {
        int i = tid - 32;
        float a = 0.5f * wgt[NQ + i];                        // wgt[1][w]
        float sv, cv; sincosf(a, &sv, &cv);
        saux[28 + i] = cv; saux[42 + i] = sv;
    }
    __syncthreads();

    // --- product state after encoding + layer-0 RY, written directly through
    // the first CNOT-ring permutation.  t = 2*tid + 512*i: bit0 wire0,
    // bits1..8 = tid, bits9..13 = i.  P(t) = ybase(tid) ^ K(i); partner ^0x3FFE.
    {
        float cr[NQ], sr[NQ];
#pragma unroll
        for (int w = 0; w < NQ; ++w) { cr[w] = saux[w]; sr[w] = saux[NQ + w]; }

        float pb = ((tid >> 0) & 1) ? sr[1] : cr[1];
#pragma unroll
        for (int w = 2; w <= 8; ++w)
            pb *= ((tid >> (w - 1)) & 1) ? sr[w] : cr[w];

        int ybase = par(tid);                                // w0: parity(2tid & 0x3FFE)
#pragma unroll
        for (int w = 1; w <= 8; ++w)
            ybase |= par(tid & ((1 << w) - 1)) << w;
        {
            int pt = par(tid);
            ybase |= (pt << 9) | (pt << 10) | (pt << 11) | (pt << 12) | (pt << 13);
        }

#pragma unroll
        for (int i = 0; i < 32; ++i) {
            float ph = ((i >> 0) & 1) ? sr[9] : cr[9];       // compile-time picks
            ph *= ((i >> 1) & 1) ? sr[10] : cr[10];
            ph *= ((i >> 2) & 1) ? sr[11] : cr[11];
            ph *= ((i >> 3) & 1) ? sr[12] : cr[12];
            ph *= ((i >> 4) & 1) ? sr[13] : cr[13];
            float pbh = pb * ph;
            const int K = par(i) | (par(i & 1) << 9) | (par(i & 3) << 10) |
                          (par(i & 7) << 11) | (par(i & 15) << 12) | (par(i) << 13);
            const int y0 = ybase ^ K;
            sbuf0[y0] = pbh * cr[0];
            sbuf0[y0 ^ 0x3FFE] = pbh * sr[0];
        }
    }
    __syncthreads();

    // --- layer 1: RY wires 0..7 (WMMA, in place) -----------------------------
    fused_lm(sbuf0, &saux[28], &saux[42], tid);
    __syncthreads();
    // --- layer 1: RY wires 8..13 + second CNOT ring -> sbuf1 -----------------
    fused_hi(sbuf0, sbuf1, &saux[28], &saux[42], tid);
    __syncthreads();

    // --- expectations --------------------------------------------------------
    {
        float tot = 0.f, a0 = 0.f, a9 = 0.f, a10 = 0.f, a11 = 0.f, a12 = 0.f, a13 = 0.f;
#pragma unroll
        for (int i = 0; i < 32; ++i) {
            const int tp = 2 * tid + 512 * i;
            v2f v = *(v2f*)(sbuf1 + tp);
            float p0 = v.x * v.x, p1 = v.y * v.y;
            float s = p0 + p1, d = p0 - p1;
            tot += s; a0 += d;
            if (i & 1)  a9  -= s; else a9  += s;
            if (i & 2)  a10 -= s; else a10 += s;
            if (i & 4)  a11 -= s; else a11 += s;
            if (i & 8)  a12 -= s; else a12 += s;
            if (i & 16) a13 -= s; else a13 += s;
        }
        float acc[NQ];
        acc[0] = a0;
#pragma unroll
        for (int w = 1; w <= 8; ++w)
            acc[w] = ((tid >> (w - 1)) & 1) ? -tot : tot;
        acc[9] = a9; acc[10] = a10; acc[11] = a11; acc[12] = a12; acc[13] = a13;

#pragma unroll
        for (int w = 0; w < NQ; ++w) {
#pragma unroll
            for (int o = 16; o > 0; o >>= 1)
                acc[w] += __shfl_xor(acc[w], o, 32);
        }
        if (lane == 0) {
#pragma unroll
            for (int w = 0; w < NQ; ++w) sred[wv * NQ + w] = acc[w];
        }
    }
    __syncthreads();

    if (tid < NQ) {
        float s = 0.f;
#pragma unroll
        for (int i = 0; i < 8; ++i) s += sred[i * NQ + tid];
        out[b * NQ + tid] = s;
    }
}

extern "C" void kernel_launch(void* const* d_in, const int* in_sizes, int n_in,
                              void* d_out, int out_size, void* d_ws, size_t ws_size,
                              hipStream_t stream) {
    const float* x   = (const float*)d_in[0];   // (B, 14) float32
    const float* wgt = (const float*)d_in[1];   // (2, 14) float32
    float* out = (float*)d_out;                 // (B, 14) float32
    const int B = in_sizes[0] / NQ;             // 512
    QuantumNeuralNetwork_73083163509199_kernel<<<B, NT, 0, stream>>>(x, wgt, out);
}